// GATGraphEncoder_40681930228168
// MI455X (gfx1250) — compile-verified
//
#include <hip/hip_runtime.h>
#include <math.h>

// ---------------------------------------------------------------------------
// Types for CDNA5 WMMA (wave32). V_WMMA_F32_16X16X4_F32:
//   A: 16x4 f32  -> 2 VGPRs/lane (v2f)
//   B:  4x16 f32 -> 2 VGPRs/lane (v2f)
//   C/D: 16x16 f32 -> 8 VGPRs/lane (v8f)
// ---------------------------------------------------------------------------
typedef float v2f __attribute__((ext_vector_type(2)));
typedef float v8f __attribute__((ext_vector_type(8)));

#define NEG_SLOPE 0.2f

// ---------------------------------------------------------------------------
// Fill a float buffer with a constant (init -inf / 0 accumulators).
// ---------------------------------------------------------------------------
__global__ void fill_kernel(float* __restrict__ p, float v, long n) {
    long i = (long)blockIdx.x * blockDim.x + threadIdx.x;
    long stride = (long)gridDim.x * blockDim.x;
    for (; i < n; i += stride) p[i] = v;
}

// ---------------------------------------------------------------------------
// H = X @ W   with X:[rows,128], W:[128,128], H:[rows,128]  (rows % 16 == 0)
// One wave computes one 16x16 tile of H via 32 chained f32 WMMAs (K=128).
// Block = 256 threads = 8 waves = one 16-row stripe across all 128 columns.
//
// A fragment (ISA 7.12.2, 32-bit A 16x4): lanes 0-15 hold M=lane, VGPR0=K0,
// VGPR1=K0+1; lanes 16-31 hold M=lane-16 with K0+2/K0+3.
// B fragment (4x16, row striped across lanes): lanes 0-15 hold N=lane with
// VGPR0=row K0, VGPR1=row K0+1; lanes 16-31 hold rows K0+2/K0+3.
// D (16x16 f32): acc[v] is element (M = v + 8*(lane>=16), N = lane&15).
// ---------------------------------------------------------------------------
__global__ void gemm128_wmma_kernel(const float* __restrict__ X,
                                    const float* __restrict__ W,
                                    float* __restrict__ Hout, int rows) {
    const int wave = threadIdx.x >> 5;
    const int lane = threadIdx.x & 31;
    const int row0 = blockIdx.x * 16;
    if (row0 + 16 > rows) return;
    const int col0 = wave * 16;
    const int half = lane >> 4;   // 0: lanes 0-15, 1: lanes 16-31
    const int l    = lane & 15;

    const float* __restrict__ xrow = X + (long)(row0 + l) * 128;

    v8f acc = {};
#pragma unroll
    for (int k0 = 0; k0 < 128; k0 += 4) {
        const int ka = k0 + half * 2;
        v2f a = *(const v2f*)(xrow + ka);                 // A[M=l][ka, ka+1]
        v2f b;
        b.x = W[(long)(ka + 0) * 128 + col0 + l];         // B[ka  ][N=l]
        b.y = W[(long)(ka + 1) * 128 + col0 + l];         // B[ka+1][N=l]
        acc = __builtin_amdgcn_wmma_f32_16x16x4_f32(
            /*neg_a=*/false, a, /*neg_b=*/false, b,
            /*c_mod=*/(short)0, acc, /*reuse_a=*/false, /*reuse_b=*/false);
    }

    const int mbase = half * 8;
#pragma unroll
    for (int v = 0; v < 8; ++v)
        Hout[(long)(row0 + mbase + v) * 128 + col0 + l] = acc[v];
}

// ---------------------------------------------------------------------------
// a_src[n,h] = dot(h[n,h,:], att_src[h,:]) ; same for a_dst. One thread/(n,h).
// ---------------------------------------------------------------------------
__global__ void att_scores_kernel(const float* __restrict__ Hm,
                                  const float* __restrict__ attS,
                                  const float* __restrict__ attD,
                                  float* __restrict__ aS, float* __restrict__ aD,
                                  int N, int H, int C) {
    int idx = blockIdx.x * blockDim.x + threadIdx.x;
    if (idx >= N * H) return;
    int n = idx / H, h = idx % H;
    const float* hp = Hm + (long)n * (H * C) + (long)h * C;
    const float* as = attS + (long)h * C;
    const float* ad = attD + (long)h * C;
    float s = 0.f, d = 0.f;
    for (int c = 0; c < C; ++c) { float v = hp[c]; s += v * as[c]; d += v * ad[c]; }
    aS[idx] = s; aD[idx] = d;
}

// float atomic max via native integer atomics (addr initialized to -inf).
__device__ __forceinline__ void atomicMaxF32(float* addr, float v) {
    if (v >= 0.f) atomicMax((int*)addr, __float_as_int(v));
    else          atomicMin((unsigned int*)addr, __float_as_uint(v));
}

// ---------------------------------------------------------------------------
// Pass 1: e = leaky_relu(a_src[src]+a_dst[dst]); segment-max into m[dst].
// ---------------------------------------------------------------------------
__global__ void edge_max_kernel(const long long* __restrict__ src,
                                const long long* __restrict__ dst,
                                const float* __restrict__ aS,
                                const float* __restrict__ aD,
                                float* __restrict__ m, int E, int H) {
    int idx = blockIdx.x * blockDim.x + threadIdx.x;
    if (idx >= E * H) return;
    int e = idx / H, h = idx % H;
    long sj = (long)src[e], di = (long)dst[e];
    float v = aS[sj * H + h] + aD[di * H + h];
    v = v > 0.f ? v : NEG_SLOPE * v;
    atomicMaxF32(&m[di * H + h], v);
}

// ---------------------------------------------------------------------------
// Pass 2: ex = exp(e - m[dst]); stash per-edge ex; segment-sum into s[dst].
// ---------------------------------------------------------------------------
__global__ void edge_exp_kernel(const long long* __restrict__ src,
                                const long long* __restrict__ dst,
                                const float* __restrict__ aS,
                                const float* __restrict__ aD,
                                const float* __restrict__ m,
                                float* __restrict__ expE,
                                float* __restrict__ sB, int E, int H) {
    int idx = blockIdx.x * blockDim.x + threadIdx.x;
    if (idx >= E * H) return;
    int e = idx / H, h = idx % H;
    long sj = (long)src[e], di = (long)dst[e];
    float v = aS[sj * H + h] + aD[di * H + h];
    v = v > 0.f ? v : NEG_SLOPE * v;
    float ex = expf(v - m[di * H + h]);
    expE[idx] = ex;
    unsafeAtomicAdd(&sB[di * H + h], ex);   // global_atomic_add_f32
}

// ---------------------------------------------------------------------------
// Pass 3: acc[dst, h, c] += h[src, h, c] * (expE / (s[dst]+eps)).
// One wave per edge; each lane moves a float4 (32 lanes x 4 = 128 feats).
// Gathers hit the 192MB L2 (h table is 51MB); atomics land in L2 as well.
// ---------------------------------------------------------------------------
__global__ void edge_scatter_kernel(const long long* __restrict__ src,
                                    const long long* __restrict__ dst,
                                    const float* __restrict__ Hm,
                                    const float* __restrict__ expE,
                                    const float* __restrict__ sB,
                                    float* __restrict__ acc, int E, int H, int C) {
    const int wave = threadIdx.x >> 5;
    const int lane = threadIdx.x & 31;
    long e = (long)blockIdx.x * 8 + wave;
    if (e >= E) return;
    long sj = (long)src[e], di = (long)dst[e];
    int feat = lane * 4;
    int h = feat / C;                               // C=32 -> lane/8 ; C=128 -> 0
    float al = expE[e * H + h] / (sB[di * H + h] + 1e-16f);
    const float4 hv = *(const float4*)(Hm + sj * 128 + feat);
    float* o = acc + di * 128 + feat;
    unsafeAtomicAdd(o + 0, hv.x * al);
    unsafeAtomicAdd(o + 1, hv.y * al);
    unsafeAtomicAdd(o + 2, hv.z * al);
    unsafeAtomicAdd(o + 3, hv.w * al);
}

// ---------------------------------------------------------------------------
// In-place x = elu(x + bias[i % C]).
// ---------------------------------------------------------------------------
__global__ void bias_elu_kernel(float* __restrict__ x, const float* __restrict__ bias,
                                long n, int C) {
    long i = (long)blockIdx.x * blockDim.x + threadIdx.x;
    long stride = (long)gridDim.x * blockDim.x;
    for (; i < n; i += stride) {
        float v = x[i] + bias[i % C];
        x[i] = v > 0.f ? v : (expf(v) - 1.f);
    }
}

// ---------------------------------------------------------------------------
static inline int cdiv(long a, long b) { return (int)((a + b - 1) / b); }

extern "C" void kernel_launch(void* const* d_in, const int* in_sizes, int n_in,
                              void* d_out, int out_size, void* d_ws, size_t ws_size,
                              hipStream_t stream) {
    const float*     x    = (const float*)d_in[0];
    const long long* a    = (const long long*)d_in[1];   // int64 [2,E]
    const float*     W1   = (const float*)d_in[2];
    const float*     as1  = (const float*)d_in[3];
    const float*     ad1  = (const float*)d_in[4];
    const float*     b1   = (const float*)d_in[5];
    const float*     W2   = (const float*)d_in[6];
    const float*     as2  = (const float*)d_in[7];
    const float*     ad2  = (const float*)d_in[8];
    const float*     b2   = (const float*)d_in[9];
    float* out = (float*)d_out;

    const int N = in_sizes[0] / 128;
    const int E = in_sizes[1] / 2;
    const long long* src = a;
    const long long* dst = a + E;

    // Workspace layout (floats)
    float* ws     = (float*)d_ws;
    float* bufH   = ws;                          // [N,128] current layer's h
    float* bufAcc = bufH   + (long)N * 128;      // [N,128] layer1 acc -> x2
    float* aS     = bufAcc + (long)N * 128;      // [N,4]
    float* aD     = aS     + (long)N * 4;        // [N,4]
    float* mB     = aD     + (long)N * 4;        // [N,4]
    float* sB     = mB     + (long)N * 4;        // [N,4]
    float* eE     = sB     + (long)N * 4;        // [E,4]

    const int FB = 2048;  // fill grid

    // ================= Layer 1: heads=4, C=32, concat =================
    gemm128_wmma_kernel<<<N / 16, 256, 0, stream>>>(x, W1, bufH, N);
    att_scores_kernel<<<cdiv((long)N * 4, 256), 256, 0, stream>>>(bufH, as1, ad1, aS, aD, N, 4, 32);
    fill_kernel<<<FB, 256, 0, stream>>>(mB, -INFINITY, (long)N * 4);
    fill_kernel<<<FB, 256, 0, stream>>>(sB, 0.f, (long)N * 4);
    fill_kernel<<<FB, 256, 0, stream>>>(bufAcc, 0.f, (long)N * 128);
    edge_max_kernel<<<cdiv((long)E * 4, 256), 256, 0, stream>>>(src, dst, aS, aD, mB, E, 4);
    edge_exp_kernel<<<cdiv((long)E * 4, 256), 256, 0, stream>>>(src, dst, aS, aD, mB, eE, sB, E, 4);
    edge_scatter_kernel<<<cdiv((long)E, 8), 256, 0, stream>>>(src, dst, bufH, eE, sB, bufAcc, E, 4, 32);
    bias_elu_kernel<<<FB, 256, 0, stream>>>(bufAcc, b1, (long)N * 128, 128);

    // ================= Layer 2: heads=1, C=128, mean(=identity) ========
    gemm128_wmma_kernel<<<N / 16, 256, 0, stream>>>(bufAcc, W2, bufH, N);
    att_scores_kernel<<<cdiv((long)N, 256), 256, 0, stream>>>(bufH, as2, ad2, aS, aD, N, 1, 128);
    fill_kernel<<<FB, 256, 0, stream>>>(mB, -INFINITY, (long)N);
    fill_kernel<<<FB, 256, 0, stream>>>(sB, 0.f, (long)N);
    fill_kernel<<<FB, 256, 0, stream>>>(out, 0.f, (long)N * 128);
    edge_max_kernel<<<cdiv((long)E, 256), 256, 0, stream>>>(src, dst, aS, aD, mB, E, 1);
    edge_exp_kernel<<<cdiv((long)E, 256), 256, 0, stream>>>(src, dst, aS, aD, mB, eE, sB, E, 1);
    edge_scatter_kernel<<<cdiv((long)E, 8), 256, 0, stream>>>(src, dst, bufH, eE, sB, out, E, 1, 128);
    bias_elu_kernel<<<FB, 256, 0, stream>>>(out, b2, (long)N * 128, 128);
}